// GraphAU_Encoder_12841952215161
// MI455X (gfx1250) — compile-verified
//
#include <hip/hip_runtime.h>

// ---------------------------------------------------------------------------
// GraphAU encoder (LightGCN-style): 3 x SpMM over 2.4M edges, D=64, then avg.
// MI455X reasoning: total working set (~144MB) fits in the 192MB L2, so the
// kernel is gather/scatter-atomic bound. We stream edge groups of 16, build
// diag(vals) in registers and use V_WMMA_F32_16X16X4_F32 (4 chained K=4
// slices = K=16) to produce the scaled 16x16 feature tiles in full f32
// precision, then scatter via global_atomic_add_f32 (64B-contiguous bursts
// per row). Edge-stream prefetch via global_prefetch_b8.
// ---------------------------------------------------------------------------

typedef __attribute__((ext_vector_type(2))) float v2f;
typedef __attribute__((ext_vector_type(8))) float v8f;

#define FEAT 64
#define HOPS_FIXED 3   // reference 'hops' is the Python constant 3

// ego = concat(user, item); acc = 0
__global__ void k_init(const float4* __restrict__ user,
                       const float4* __restrict__ item,
                       float4* __restrict__ ego,
                       float4* __restrict__ acc,
                       int nUser4, int nTot4) {
    for (int i = blockIdx.x * blockDim.x + threadIdx.x; i < nTot4;
         i += gridDim.x * blockDim.x) {
        float4 v = (i < nUser4) ? user[i] : item[i - nUser4];
        ego[i] = v;
        acc[i] = make_float4(0.f, 0.f, 0.f, 0.f);
    }
}

__global__ void k_zero(float4* __restrict__ p, int n4) {
    for (int i = blockIdx.x * blockDim.x + threadIdx.x; i < n4;
         i += gridDim.x * blockDim.x)
        p[i] = make_float4(0.f, 0.f, 0.f, 0.f);
}

// acc = (acc + next) * scale   (scale = 1 for inner hops, 1/HOPS on the last)
__global__ void k_accum(float4* __restrict__ acc,
                        const float4* __restrict__ next,
                        float scale, int n4) {
    for (int i = blockIdx.x * blockDim.x + threadIdx.x; i < n4;
         i += gridDim.x * blockDim.x) {
        float4 a = acc[i], b = next[i];
        acc[i] = make_float4((a.x + b.x) * scale, (a.y + b.y) * scale,
                             (a.z + b.z) * scale, (a.w + b.w) * scale);
    }
}

// SpMM: y[row] += val * x[col], processed 16 edges per wave-iteration.
// For each 16-edge group and each 16-feature tile:
//   D(16x16) = diag(val_0..15) * B,  B[k, n] = x[col_k][n0+n]
// computed as 4 chained V_WMMA_F32_16X16X4_F32 (K slices of 4).
__global__ void k_spmm_wmma(const int*   __restrict__ rows,
                            const int*   __restrict__ cols,
                            const float* __restrict__ vals,
                            const float* __restrict__ x,
                            float*       __restrict__ y,
                            int nGroups) {
    const int lane = threadIdx.x & 31;
    const int m    = lane & 15;   // A-row / B-col / D-col index for this lane
    const int hi   = lane >> 4;   // which half of the wave (K/M split)
    const int wavesPerBlock = blockDim.x >> 5;
    const int wave  = blockIdx.x * wavesPerBlock + (threadIdx.x >> 5);
    const int nWave = gridDim.x * wavesPerBlock;

    for (int g = wave; g < nGroups; g += nWave) {   // wave-uniform loop
        const int base = g << 4;
        // every lane holds edge (base + m); halves are redundant (L0/L2 hit)
        const int   row = rows[base + m];
        const int   col = cols[base + m];
        const float val = vals[base + m];

        // prefetch the next group's edge stream (global_prefetch_b8)
        if (g + nWave < nGroups) {
            const int nb = (g + nWave) << 4;
            __builtin_prefetch(&rows[nb], 0, 1);
            __builtin_prefetch(&cols[nb], 0, 1);
            __builtin_prefetch(&vals[nb], 0, 1);
        }

#pragma unroll
        for (int t = 0; t < 4; ++t) {          // four 16-wide feature tiles
            const int n0 = t << 4;
            v8f c = {};                        // C = 0
#pragma unroll
            for (int kb = 0; kb < 4; ++kb) {   // K = 16 as 4 slices of 4
                const int k0 = (kb << 2) + (hi << 1); // global K of a.x / b.x
                // A slice, 16x4 layout (ISA 7.12.2): lane m, v0=K(2hi), v1=K(2hi+1)
                v2f a, b;
                a.x = (m == k0    ) ? val : 0.0f;
                a.y = (m == k0 + 1) ? val : 0.0f;
                // B slice, 4x16 layout: v0 = row K(2hi), v1 = row K(2hi+1), N = lane&15
                const int c0 = __shfl(col, k0,     32);
                const int c1 = __shfl(col, k0 + 1, 32);
                b.x = x[c0 * FEAT + n0 + m];
                b.y = x[c1 * FEAT + n0 + m];
                c = __builtin_amdgcn_wmma_f32_16x16x4_f32(
                        false, a, false, b, (short)0, c, false, false);
            }
            // D layout: VGPR j -> M = j + 8*hi, N = m. Scatter 64B bursts/row.
#pragma unroll
            for (int j = 0; j < 8; ++j) {
                const int r = __shfl(row, j + (hi << 3), 32);
                atomicAdd(&y[r * FEAT + n0 + m], c[j]);
            }
        }
    }
}

// Scalar tail for E not a multiple of 16 (unused for E=2.4M, kept for safety).
__global__ void k_spmm_tail(const int*   __restrict__ rows,
                            const int*   __restrict__ cols,
                            const float* __restrict__ vals,
                            const float* __restrict__ x,
                            float*       __restrict__ y,
                            int eBase, int eEnd) {
    int tid = blockIdx.x * blockDim.x + threadIdx.x;
    int e   = eBase + (tid >> 6);          // 64 threads per edge
    int d   = tid & 63;
    if (e < eEnd)
        atomicAdd(&y[rows[e] * FEAT + d], vals[e] * x[cols[e] * FEAT + d]);
}

extern "C" void kernel_launch(void* const* d_in, const int* in_sizes, int n_in,
                              void* d_out, int out_size, void* d_ws, size_t ws_size,
                              hipStream_t stream) {
    const float* user = (const float*)d_in[0];
    const float* item = (const float*)d_in[1];
    const int*   rows = (const int*)d_in[2];
    const int*   cols = (const int*)d_in[3];
    const float* vals = (const float*)d_in[4];

    const int nUserF = in_sizes[0];            // N_USERS * 64
    const int nTotF  = in_sizes[0] + in_sizes[1];
    const int E      = in_sizes[2];
    const int n4     = nTotF / 4;

    float* ego = (float*)d_ws;                 // ping
    float* nxt = (float*)d_ws + nTotF;         // pong
    float* acc = (float*)d_out;                // accumulator lives in d_out

    const int EW_THREADS = 256;
    int ewBlocks = (n4 + EW_THREADS - 1) / EW_THREADS;
    if (ewBlocks > 4096) ewBlocks = 4096;      // grid-stride

    k_init<<<ewBlocks, EW_THREADS, 0, stream>>>(
        (const float4*)user, (const float4*)item,
        (float4*)ego, (float4*)acc, nUserF / 4, n4);

    const int nGroups  = E >> 4;
    const int tailBase = nGroups << 4;
    const int SP_THREADS = 256;                // 8 waves/block
    int spBlocks = 1200;                       // 9600 waves over 150k groups

    for (int h = 0; h < HOPS_FIXED; ++h) {
        k_zero<<<ewBlocks, EW_THREADS, 0, stream>>>((float4*)nxt, n4);
        k_spmm_wmma<<<spBlocks, SP_THREADS, 0, stream>>>(
            rows, cols, vals, ego, nxt, nGroups);
        if (tailBase < E) {
            int tailThreads = (E - tailBase) * FEAT;
            k_spmm_tail<<<(tailThreads + 255) / 256, 256, 0, stream>>>(
                rows, cols, vals, ego, nxt, tailBase, E);
        }
        const float scale = (h == HOPS_FIXED - 1) ? (1.0f / HOPS_FIXED) : 1.0f;
        k_accum<<<ewBlocks, EW_THREADS, 0, stream>>>(
            (float4*)acc, (const float4*)nxt, scale, n4);
        float* tmp = ego; ego = nxt; nxt = tmp;   // deterministic host swap
    }
}